// LightGCN_17755394802266
// MI455X (gfx1250) — compile-verified
//
#include <hip/hip_runtime.h>
#include <hip/hip_bf16.h>

#define DIM 64

typedef __attribute__((ext_vector_type(2))) float v2f;
typedef __attribute__((ext_vector_type(8))) float v8f;

// ---------------- degree / normalization ----------------

__global__ void zero_deg_kernel(unsigned* __restrict__ deg, int N) {
    int i = blockIdx.x * blockDim.x + threadIdx.x;
    if (i < N) deg[i] = 0u;
}

__global__ void count_deg_kernel(const int* __restrict__ row, unsigned* __restrict__ deg, int E) {
    int e = blockIdx.x * blockDim.x + threadIdx.x;
    if (e < E) atomicAdd(&deg[row[e]], 1u);
}

// deg and dinv alias the same buffer: read u32 count, write fp32 rsqrt.
__global__ void dinv_kernel(unsigned* __restrict__ deg, float* __restrict__ dinv, int N) {
    int i = blockIdx.x * blockDim.x + threadIdx.x;
    if (i < N) {
        unsigned d = deg[i] + 1u;            // +1 for the self-loop edge
        dinv[i] = rsqrtf((float)d);          // d >= 1 always
    }
}

// ---------------- embedding init (cur = acc = emb) ----------------

__global__ void init_embed_kernel(const float* __restrict__ uw, const float* __restrict__ iw,
                                  float* __restrict__ cur, float* __restrict__ acc,
                                  int U, int N) {
    long t = (long)blockIdx.x * blockDim.x + threadIdx.x;
    long total = (long)N * (DIM / 4);
    if (t >= total) return;
    int n  = (int)(t >> 4);
    int ch = (int)(t & 15);
    const float4* src = (n < U) ? (const float4*)(uw + (size_t)n * DIM)
                                : (const float4*)(iw + (size_t)(n - U) * DIM);
    float4 v = src[ch];
    ((float4*)(cur + (size_t)n * DIM))[ch] = v;
    ((float4*)(acc + (size_t)n * DIM))[ch] = v;
}

// ---------------- per-layer: nxt = cur * dinv^2 (zero-init + self-loop fused) ----------------

__global__ void selfloop_kernel(const float* __restrict__ cur, const float* __restrict__ dinv,
                                float* __restrict__ nxt, int N) {
    long t = (long)blockIdx.x * blockDim.x + threadIdx.x;
    long total = (long)N * (DIM / 4);
    if (t >= total) return;
    int n  = (int)(t >> 4);
    int ch = (int)(t & 15);
    float w = dinv[n];
    float s = w * w;
    float4 v = ((const float4*)(cur + (size_t)n * DIM))[ch];
    v.x *= s; v.y *= s; v.z *= s; v.w *= s;
    ((float4*)(nxt + (size_t)n * DIM))[ch] = v;
}

// ---------------- per-layer: scatter-add over edges ----------------
// 16 lanes per edge; each lane: one coalesced float4 gather + 4 no-return fp32 atomics.

__global__ void scatter_kernel(const int* __restrict__ row, const int* __restrict__ col,
                               const float* __restrict__ dinv, const float* __restrict__ cur,
                               float* __restrict__ nxt, int E) {
    long t = (long)blockIdx.x * blockDim.x + threadIdx.x;
    if (t >= (long)E * 16) return;
    int e = (int)(t >> 4);
    int l = (int)(t & 15);
    int r = row[e];
    int c = col[e];
    float norm = dinv[r] * dinv[c];
    float4 v = *(const float4*)(cur + (size_t)c * DIM + l * 4);
    float* dst = nxt + (size_t)r * DIM + l * 4;
    unsafeAtomicAdd(dst + 0, v.x * norm);
    unsafeAtomicAdd(dst + 1, v.y * norm);
    unsafeAtomicAdd(dst + 2, v.z * norm);
    unsafeAtomicAdd(dst + 3, v.w * norm);
}

// ---------------- per-layer: acc += nxt ----------------

__global__ void acc_add_kernel(float* __restrict__ acc, const float* __restrict__ nxt, long total4) {
    long t = (long)blockIdx.x * blockDim.x + threadIdx.x;
    if (t >= total4) return;
    float4 a = ((float4*)acc)[t];
    float4 b = ((const float4*)nxt)[t];
    a.x += b.x; a.y += b.y; a.z += b.z; a.w += b.w;
    ((float4*)acc)[t] = a;
}

// ---------------- finalize: acc *= 0.25 via V_WMMA_F32_16X16X4_F32 ----------------
// Each wave scales one 16x16 fp32 tile as D = sum_t A_t(16x4) x B_t(4x16) with
// B = 0.25 * I (exact: 0.25 is a power of two; off-diagonal terms are exact zeros).
// A layout (16x4 f32): VGPR v, lane group g=lane>>4 -> K = 2g + v, M = lane&15.
// B layout (4x16 f32): VGPR v, lane group g        -> K = 2g + v, N = lane&15.
// C/D layout: VGPR r -> M = r + 8g, N = lane&15.

__global__ void finalize_wmma_kernel(float* __restrict__ acc, int nWaves) {
    int wave = (blockIdx.x * blockDim.x + threadIdx.x) >> 5;
    int lane = threadIdx.x & 31;
    if (wave >= nWaves) return;           // wave-uniform guard: EXEC stays all-1s past here
    int rowTile = wave >> 2;              // 4 column tiles per 16-row tile (DIM=64)
    int colTile = wave & 3;
    int m0 = rowTile << 4;
    int c0 = colTile << 4;
    int half = lane >> 4;
    int lid  = lane & 15;

    const float* arow = acc + (size_t)(m0 + lid) * DIM + c0;
    v8f d = {0.f, 0.f, 0.f, 0.f, 0.f, 0.f, 0.f, 0.f};
#pragma unroll
    for (int t = 0; t < 4; ++t) {
        int k0 = 4 * t + 2 * half;
        v2f a;
        a.x = arow[k0];
        a.y = arow[k0 + 1];
        v2f b;
        b.x = (lid == k0)     ? 0.25f : 0.0f;
        b.y = (lid == k0 + 1) ? 0.25f : 0.0f;
        // 8 args: (neg_a, A, neg_b, B, c_mod, C, reuse_a, reuse_b)
        d = __builtin_amdgcn_wmma_f32_16x16x4_f32(false, a, false, b, (short)0, d, false, false);
    }
    float* out = acc + (size_t)(m0 + 8 * half) * DIM + c0 + lid;
#pragma unroll
    for (int r = 0; r < 8; ++r) out[(size_t)r * DIM] = d[r];
}

__global__ void scale_tail_kernel(float* __restrict__ acc, long start, long total) {
    long t = start + (long)blockIdx.x * blockDim.x + threadIdx.x;
    if (t < total) acc[t] *= 0.25f;
}

// ---------------- launcher ----------------

extern "C" void kernel_launch(void* const* d_in, const int* in_sizes, int n_in,
                              void* d_out, int out_size, void* d_ws, size_t ws_size,
                              hipStream_t stream) {
    const int*   edges = (const int*)d_in[0];   // [2, E] int32
    const float* uw    = (const float*)d_in[1]; // [U, 64] f32
    const float* iw    = (const float*)d_in[2]; // [I, 64] f32

    const int E = in_sizes[0] / 2;
    const int U = in_sizes[1] / DIM;
    const int I = in_sizes[2] / DIM;
    const int N = U + I;

    const int* row = edges;
    const int* col = edges + E;

    char* ws = (char*)d_ws;
    unsigned* deg  = (unsigned*)ws;            // N * 4 bytes, aliased with dinv
    float*    dinv = (float*)ws;
    const size_t offA     = (size_t)1 << 20;   // 1 MB
    const size_t bufBytes = (size_t)N * DIM * sizeof(float);  // 38.4 MB
    float* bufA = (float*)(ws + offA);
    float* bufB = (float*)(ws + offA + bufBytes);
    float* acc  = (float*)d_out;

    const int B = 256;
    zero_deg_kernel<<<(N + B - 1) / B, B, 0, stream>>>(deg, N);
    count_deg_kernel<<<(E + B - 1) / B, B, 0, stream>>>(row, deg, E);
    dinv_kernel<<<(N + B - 1) / B, B, 0, stream>>>(deg, dinv, N);

    const long vec4Total = (long)N * (DIM / 4);   // N*16 float4 slots
    init_embed_kernel<<<(int)((vec4Total + B - 1) / B), B, 0, stream>>>(uw, iw, bufA, acc, U, N);

    float* cur = bufA;
    float* nxt = bufB;
    for (int layer = 0; layer < 3; ++layer) {
        selfloop_kernel<<<(int)((vec4Total + B - 1) / B), B, 0, stream>>>(cur, dinv, nxt, N);
        const long st = (long)E * 16;
        scatter_kernel<<<(int)((st + B - 1) / B), B, 0, stream>>>(row, col, dinv, cur, nxt, E);
        acc_add_kernel<<<(int)((vec4Total + B - 1) / B), B, 0, stream>>>(acc, nxt, vec4Total);
        float* tmp = cur; cur = nxt; nxt = tmp;
    }

    const int nRowTiles = N >> 4;        // 9375 for N=150000 (exact)
    const int nWaves    = nRowTiles * 4; // one wave per 16x16 tile
    if (nWaves > 0) {
        const int threads = 128;         // 4 wave32 per block
        const int blocks  = (nWaves * 32 + threads - 1) / threads;
        finalize_wmma_kernel<<<blocks, threads, 0, stream>>>(acc, nWaves);
    }
    const long tailStart  = (long)(nRowTiles << 4) * DIM;
    const long totalElems = (long)N * DIM;
    if (tailStart < totalElems) {
        const long tailCount = totalElems - tailStart;
        scale_tail_kernel<<<(int)((tailCount + B - 1) / B), B, 0, stream>>>(acc, tailStart, totalElems);
    }
}